// SinkhornLayer_4398046511616
// MI455X (gfx1250) — compile-verified
//
#include <hip/hip_runtime.h>
#include <hip/hip_bf16.h>

#define N 8192
#define D 64
#define NT (N / 16)          // 512 tiles per dimension
#define RCHUNKS 16
#define RROWS (N / RCHUNKS)  // 512 rows per column-matvec chunk
#define MAX_ITER 100

typedef __attribute__((ext_vector_type(2))) float v2f;
typedef __attribute__((ext_vector_type(8))) float v8f;

// ---------------------------------------------------------------------------
// Row squared norms of x and y: xn[i] = sum_k x[i][k]^2, yn likewise.
// grid 64 x 256 covers 2*8192 rows.
// ---------------------------------------------------------------------------
__global__ __launch_bounds__(256) void prep_norms(const float* __restrict__ x,
                                                  const float* __restrict__ y,
                                                  float* __restrict__ xn,
                                                  float* __restrict__ yn) {
  int gid = blockIdx.x * 256 + threadIdx.x;
  int row = (gid < N) ? gid : (gid - N);
  const float* src = (gid < N) ? x : y;
  float s = 0.f;
#pragma unroll
  for (int k = 0; k < D; ++k) {
    float t = src[(size_t)row * D + k];
    s += t * t;
  }
  if (gid < N) xn[row] = s; else yn[row] = s;
}

__global__ __launch_bounds__(256) void init_u(float* __restrict__ u) {
  int gid = blockIdx.x * 256 + threadIdx.x;
  u[gid] = 1.0f / (float)N;
}

// ---------------------------------------------------------------------------
// K = exp(-10 * sqrt(clip(||x||^2 + ||y||^2 - 2 x.y, 1e-12)))
// One wave per 16x16 tile; fp32 WMMA 16x16x4, 16 steps over d=64.
// A (16x4 fp32): lane L (m=L&15, half=L>>4) holds {A[m][k0+2h], A[m][k0+2h+1]}
// B (4x16 fp32): same pattern with n=L&15 (B[k][n] = y[n][k]).
// D (16x16 fp32): VGPR r, lane L -> row r + 8*half, col L&15.
// ---------------------------------------------------------------------------
__global__ __launch_bounds__(256) void genK(const float* __restrict__ x,
                                            const float* __restrict__ y,
                                            const float* __restrict__ xn,
                                            const float* __restrict__ yn,
                                            float* __restrict__ K) {
  int wave = threadIdx.x >> 5;
  int lane = threadIdx.x & 31;
  int tile = blockIdx.x * 8 + wave;       // 512*512 tiles / 8 waves per block
  int ti = tile / NT;
  int tj = tile % NT;
  int m = lane & 15;
  int half = lane >> 4;

  const float* xr = x + (size_t)(ti * 16 + m) * D;  // A row for this lane
  const float* yr = y + (size_t)(tj * 16 + m) * D;  // B column (y row)

  v8f acc = {0.f, 0.f, 0.f, 0.f, 0.f, 0.f, 0.f, 0.f};
#pragma unroll
  for (int kc = 0; kc < D / 4; ++kc) {
    int kb = kc * 4 + half * 2;
    v2f a = *(const v2f*)(xr + kb);
    v2f b = *(const v2f*)(yr + kb);
    acc = __builtin_amdgcn_wmma_f32_16x16x4_f32(
        /*neg_a=*/false, a, /*neg_b=*/false, b,
        /*c_mod=*/(short)0, acc, /*reuse_a=*/false, /*reuse_b=*/false);
  }

  int ci = tj * 16 + m;
  float ynn = yn[ci];
#pragma unroll
  for (int r = 0; r < 8; ++r) {
    int ri = ti * 16 + half * 8 + r;
    float s = xn[ri] + ynn - 2.0f * acc[r];
    s = fmaxf(s, 1e-12f);
    float c = sqrtf(s);
    K[(size_t)ri * N + ci] = __expf(-10.0f * c);   // exp(-C / 0.1)
  }
}

// ---------------------------------------------------------------------------
// Column matvec partials: partial[rc][j] = sum_{i in chunk rc} K[i][j] * u[i]
// grid (32, 16), block 256; thread-per-column, coalesced row streaming.
// 8 independent accumulator chains to keep loads in flight.
// ---------------------------------------------------------------------------
__global__ __launch_bounds__(256) void colmv(const float* __restrict__ K,
                                             const float* __restrict__ u,
                                             float* __restrict__ partial) {
  int j = blockIdx.x * 256 + threadIdx.x;
  int i0 = blockIdx.y * RROWS;
  float s0 = 0.f, s1 = 0.f, s2 = 0.f, s3 = 0.f;
  float s4 = 0.f, s5 = 0.f, s6 = 0.f, s7 = 0.f;
  for (int i = 0; i < RROWS; i += 8) {
    const float* Kp = K + (size_t)(i0 + i) * N + j;
    const float* up = u + i0 + i;
    s0 += Kp[0 * (size_t)N] * up[0];
    s1 += Kp[1 * (size_t)N] * up[1];
    s2 += Kp[2 * (size_t)N] * up[2];
    s3 += Kp[3 * (size_t)N] * up[3];
    s4 += Kp[4 * (size_t)N] * up[4];
    s5 += Kp[5 * (size_t)N] * up[5];
    s6 += Kp[6 * (size_t)N] * up[6];
    s7 += Kp[7 * (size_t)N] * up[7];
  }
  partial[(size_t)blockIdx.y * N + j] =
      ((s0 + s1) + (s2 + s3)) + ((s4 + s5) + (s6 + s7));
}

// v[j] = b_j / sum_rc partial[rc][j]     (b_j = 1/m)
__global__ __launch_bounds__(256) void reddiv(const float* __restrict__ partial,
                                              float* __restrict__ v) {
  int j = blockIdx.x * 256 + threadIdx.x;
  float s = 0.f;
#pragma unroll
  for (int rc = 0; rc < RCHUNKS; ++rc) s += partial[(size_t)rc * N + j];
  v[j] = (1.0f / (float)N) / s;
}

// ---------------------------------------------------------------------------
// Row matvec + divide: u[i] = a_i / sum_j K[i][j] * v[j]
// Block-per-row, float4 coalesced loads, LDS tree reduce.
// ---------------------------------------------------------------------------
__global__ __launch_bounds__(256) void rowmv(const float* __restrict__ K,
                                             const float* __restrict__ v,
                                             float* __restrict__ u) {
  int i = blockIdx.x;
  const float4* Kr = (const float4*)(K + (size_t)i * N);
  const float4* V4 = (const float4*)v;
  float s = 0.f;
#pragma unroll
  for (int c = 0; c < N / 1024; ++c) {  // 8 chunks of 1024 cols
    int idx = c * 256 + threadIdx.x;
    float4 k = Kr[idx];
    float4 vv = V4[idx];
    s += k.x * vv.x + k.y * vv.y + k.z * vv.z + k.w * vv.w;
  }
  __shared__ float red[256];
  red[threadIdx.x] = s;
  __syncthreads();
  for (int off = 128; off > 0; off >>= 1) {
    if (threadIdx.x < off) red[threadIdx.x] += red[threadIdx.x + off];
    __syncthreads();
  }
  if (threadIdx.x == 0) u[i] = (1.0f / (float)N) / red[0];
}

// pi[i][j] = u[i] * K[i][j] * v[j], in place on d_out
__global__ __launch_bounds__(256) void pi_scale(float* __restrict__ K,
                                                const float* __restrict__ u,
                                                const float* __restrict__ v) {
  int i = blockIdx.x;
  float ui = u[i];
  float4* Kr = (float4*)(K + (size_t)i * N);
  const float4* V4 = (const float4*)v;
#pragma unroll
  for (int c = 0; c < N / 1024; ++c) {
    int idx = c * 256 + threadIdx.x;
    float4 k = Kr[idx];
    float4 vv = V4[idx];
    k.x *= ui * vv.x;
    k.y *= ui * vv.y;
    k.z *= ui * vv.z;
    k.w *= ui * vv.w;
    Kr[idx] = k;
  }
}

extern "C" void kernel_launch(void* const* d_in, const int* in_sizes, int n_in,
                              void* d_out, int out_size, void* d_ws, size_t ws_size,
                              hipStream_t stream) {
  const float* x = (const float*)d_in[0];   // [8192, 64] fp32
  const float* y = (const float*)d_in[1];   // [8192, 64] fp32
  float* K = (float*)d_out;                 // K lives in d_out, rescaled to pi at end

  float* ws = (float*)d_ws;                 // ~640 KB used
  float* u = ws;                            // [N]
  float* v = ws + N;                        // [N]
  float* xn = ws + 2 * N;                   // [N]
  float* yn = ws + 3 * N;                   // [N]
  float* partial = ws + 4 * N;              // [RCHUNKS, N]

  prep_norms<<<64, 256, 0, stream>>>(x, y, xn, yn);
  init_u<<<N / 256, 256, 0, stream>>>(u);
  genK<<<(NT * NT) / 8, 256, 0, stream>>>(x, y, xn, yn, K);

  for (int it = 0; it < MAX_ITER; ++it) {
    colmv<<<dim3(N / 256, RCHUNKS), 256, 0, stream>>>(K, u, partial);
    reddiv<<<N / 256, 256, 0, stream>>>(partial, v);
    rowmv<<<N, 256, 0, stream>>>(K, v, u);
  }
  // final v = b / (K^T u), then pi = u * K * v
  colmv<<<dim3(N / 256, RCHUNKS), 256, 0, stream>>>(K, u, partial);
  reddiv<<<N / 256, 256, 0, stream>>>(partial, v);
  pi_scale<<<N, 256, 0, stream>>>(K, u, v);
}